// CITADEL_15118284882566
// MI455X (gfx1250) — compile-verified
//
#include <hip/hip_runtime.h>

typedef __attribute__((ext_vector_type(2))) float v2f;
typedef __attribute__((ext_vector_type(8))) float v8f;

#define NB   512   // batches
#define LQ   32
#define LD   512
#define KD   5
#define DD   128
#define SROW 516   // padded LDS row stride (516 mod 64 == 4 -> conflict-free tile stores)

__global__ __launch_bounds__(256)
void citadel_colbert_moe_kernel(
    const float* __restrict__ q_repr,   // [B,32,128]
    const float* __restrict__ q_w,      // [B,32,1]
    const int*   __restrict__ q_ids,    // [B,32,1]
    const float* __restrict__ q_cls,    // [B,128]
    const float* __restrict__ d_repr,   // [B,512,128]
    const float* __restrict__ d_w,      // [B,512,5]
    const int*   __restrict__ d_ids,    // [B,512,5]
    const float* __restrict__ d_cls,    // [B,128]
    float* __restrict__ out)            // [B]
{
    __shared__ float sims[LQ * SROW];   // 66 KB: sims[q][ld]
    __shared__ float wavemax[8 * LQ];
    __shared__ int   qid_s[LQ];
    __shared__ float qw_s[LQ];

    const int tid  = threadIdx.x;
    const int lane = tid & 31;
    const int wave = tid >> 5;
    const int b    = blockIdx.x;

    if (tid < LQ) {
        qid_s[tid] = q_ids[b * LQ + tid];
        qw_s[tid]  = q_w[b * LQ + tid];
    }

    const float* qb = q_repr + (size_t)b * LQ * DD;
    const float* db = d_repr + (size_t)b * LD * DD;

    // --- Phase 1: sims = Q (32x128) * D^T (128x512) via V_WMMA_F32_16X16X4_F32 ---
    // 32-bit A 16x4 layout: lanes 0-15 hold K=0,1 ; lanes 16-31 hold K=2,3 (M = lane&15).
    // B 4x16 assumed mirror layout: lanes 0-15 hold rows K=0,1 ; lanes 16-31 K=2,3 (N = lane&15).
    const int lm   = lane & 15;
    const int hs   = lane >> 4;      // half-wave select
    const int kofs = hs * 2;         // K offset within the 4-step

    const float* qrow0 = qb + (size_t)lm * DD;         // M tile 0 (rows 0..15)
    const float* qrow1 = qb + (size_t)(16 + lm) * DD;  // M tile 1 (rows 16..31)

    for (int nt = wave; nt < LD / 16; nt += 8) {       // 4 column tiles per wave
        const int ncol = nt * 16 + lm;
        const float* drow = db + (size_t)ncol * DD;
        v8f c0 = {0.f, 0.f, 0.f, 0.f, 0.f, 0.f, 0.f, 0.f};
        v8f c1 = c0;
        #pragma unroll 4
        for (int k0 = 0; k0 < DD; k0 += 4) {
            v2f a0 = *(const v2f*)(qrow0 + k0 + kofs);
            v2f a1 = *(const v2f*)(qrow1 + k0 + kofs);
            v2f bf = *(const v2f*)(drow  + k0 + kofs);
            // 8 args: (neg_a, A, neg_b, B, c_mod, C, reuse_a, reuse_b)
            c0 = __builtin_amdgcn_wmma_f32_16x16x4_f32(false, a0, false, bf,
                                                       (short)0, c0, false, false);
            c1 = __builtin_amdgcn_wmma_f32_16x16x4_f32(false, a1, false, bf,
                                                       (short)0, c1, false, false);
        }
        // C/D layout: VGPR r, lanes 0-15 -> M=r ; lanes 16-31 -> M=r+8 ; N = lane&15
        #pragma unroll
        for (int r = 0; r < 8; ++r) {
            const int q = r + hs * 8;
            sims[q * SROW + nt * 16 + lm]        = c0[r];
            sims[(16 + q) * SROW + nt * 16 + lm] = c1[r];
        }
    }
    __syncthreads();

    // --- Phase 2: masked weighted max over (ld, kd) per q ---
    float pm[LQ];
    #pragma unroll
    for (int q = 0; q < LQ; ++q) pm[q] = -__builtin_inff();

    for (int ld = tid; ld < LD; ld += 256) {           // 2 columns per thread
        const int*   idp = d_ids + ((size_t)b * LD + ld) * KD;
        const float* wp  = d_w   + ((size_t)b * LD + ld) * KD;
        int   did[KD];
        float dwv[KD];
        #pragma unroll
        for (int kd = 0; kd < KD; ++kd) { did[kd] = idp[kd]; dwv[kd] = wp[kd]; }
        #pragma unroll
        for (int q = 0; q < LQ; ++q) {
            const float s  = sims[q * SROW + ld];
            const int   qi = qid_s[q];
            float m = pm[q];
            #pragma unroll
            for (int kd = 0; kd < KD; ++kd) {
                const float v = (did[kd] == qi) ? s * dwv[kd] : 0.0f;  // masked -> exact 0
                m = fmaxf(m, v);
            }
            pm[q] = m;
        }
    }

    // wave32 butterfly max-reduce per q
    #pragma unroll
    for (int q = 0; q < LQ; ++q) {
        float m = pm[q];
        #pragma unroll
        for (int off = 16; off > 0; off >>= 1)
            m = fmaxf(m, __shfl_xor(m, off, 32));
        if (lane == 0) wavemax[wave * LQ + q] = m;
    }
    __syncthreads();

    // --- Phase 3: combine waves, apply qw (>=0 so max commutes), add cls dot ---
    if (wave == 0) {
        float m = wavemax[lane];                       // lane == q
        #pragma unroll
        for (int w = 1; w < 8; ++w) m = fmaxf(m, wavemax[w * LQ + lane]);
        float acc = qw_s[lane] * m;

        const float* qc = q_cls + (size_t)b * DD + lane * 4;
        const float* dc = d_cls + (size_t)b * DD + lane * 4;
        #pragma unroll
        for (int i = 0; i < 4; ++i) acc += qc[i] * dc[i];

        #pragma unroll
        for (int off = 16; off > 0; off >>= 1)
            acc += __shfl_xor(acc, off, 32);
        if (lane == 0) out[b] = acc;
    }
}

extern "C" void kernel_launch(void* const* d_in, const int* in_sizes, int n_in,
                              void* d_out, int out_size, void* d_ws, size_t ws_size,
                              hipStream_t stream) {
    const float* q_repr = (const float*)d_in[0];
    const float* q_w    = (const float*)d_in[1];
    const int*   q_ids  = (const int*)d_in[2];
    const float* q_cls  = (const float*)d_in[3];
    const float* d_repr = (const float*)d_in[4];
    const float* d_w    = (const float*)d_in[5];
    const int*   d_ids  = (const int*)d_in[6];
    const float* d_cls  = (const float*)d_in[7];
    float* out = (float*)d_out;
    (void)in_sizes; (void)n_in; (void)out_size; (void)d_ws; (void)ws_size;

    hipLaunchKernelGGL(citadel_colbert_moe_kernel, dim3(NB), dim3(256), 0, stream,
                       q_repr, q_w, q_ids, q_cls, d_repr, d_w, d_ids, d_cls, out);
}